// MaskedAttentionPool_13314398617966
// MI455X (gfx1250) — compile-verified
//
#include <hip/hip_runtime.h>
#include <math.h>

typedef __attribute__((ext_vector_type(2))) float v2f;
typedef __attribute__((ext_vector_type(8))) float v8f;

#define BATCH   32
#define NTOK    4096
#define DIM     1024
#define NEGVAL  (-1e9f)
#define SCALE   0.03125f          /* 1/sqrt(1024) */

#define CHUNK   32                /* token rows resident in LDS at once      */
#define CPB     8                 /* chunks processed sequentially per block */
#define ROWS_PER_BLOCK (CHUNK * CPB)            /* 256 */
#define NBLK    (NTOK / ROWS_PER_BLOCK)         /* 16 blocks per batch */
#define LDSS    1036              /* padded tile row stride (floats):
                                     - 1036*4 % 16 == 0 -> b128-aligned rows
                                     - r*1036 % 64 distinct for r=0..15 ->
                                       conflict-free A-fragment ds reads     */

// ---------------------------------------------------------------------------
// Fused kernel: single HBM pass over tokens (~512 MB total -> ~22 us floor).
// Per 32-row chunk kept in LDS:
//   1) coalesced global->LDS stage (4KB rows, b128)
//   2) WMMA f32 16x16x4 scores: 8 waves = 2 row-tiles x 4 K-quarters,
//      B operand = query[k] broadcast across all 16 columns
//   3) mask + scale, online-softmax rescale of running (m, Z, S)
//   4) weighted accumulation of the SAME LDS tile into per-thread float4
// Emits per-block partial: header (m, Z, S) + 1024-float accumulator.
// ---------------------------------------------------------------------------
__global__ __launch_bounds__(256)
void mp_fused(const float* __restrict__ tokens, const float* __restrict__ mask,
              const float* __restrict__ query,
              float* __restrict__ hdr, float* __restrict__ pacc) {
    __shared__ float tile[CHUNK * LDSS];   // ~132.6 KB
    __shared__ float q[DIM];               // 4 KB
    __shared__ float psc[8 * 16];          // per-wave partial scores
    __shared__ float sc[CHUNK];
    __shared__ float msk[CHUNK];
    __shared__ float wv[CHUNK];

    const int t    = threadIdx.x;
    const int wave = t >> 5;
    const int lane = t & 31;
    const int m16  = lane & 15;            // A/B fragment row
    const int hi   = lane >> 4;            // 0 -> K=0,1 ; 1 -> K=2,3
    const int tw   = wave & 1;             // which 16-row tile of the chunk
    const int kq   = wave >> 1;            // which 256-wide K quarter
    const long long rowsBase =
        (long long)(blockIdx.x / NBLK) * NTOK +
        (long long)(blockIdx.x % NBLK) * ROWS_PER_BLOCK;

    for (int i = t; i < DIM; i += 256) q[i] = query[i];

    float  m_run = -INFINITY, Z_run = 0.f, S_run = 0.f;
    float4 acc = {0.f, 0.f, 0.f, 0.f};

    for (int c = 0; c < CPB; ++c) {
        const long long r0 = rowsBase + c * CHUNK;

        __syncthreads();   // previous chunk's pool reads done before overwrite
        // --- 1) stage 32 x 1024 tile; thread t owns column 4t of every row
        #pragma unroll 4
        for (int r = 0; r < CHUNK; ++r) {
            const float4 v =
                *(const float4*)&tokens[(r0 + r) * (long long)DIM + 4 * t];
            *(float4*)&tile[r * LDSS + 4 * t] = v;
        }
        __syncthreads();

        // --- 2) WMMA scores for this wave's (row-tile, K-quarter)
        v8f cacc = {0.f, 0.f, 0.f, 0.f, 0.f, 0.f, 0.f, 0.f};
        const int kbase = kq * 256;
        #pragma unroll 8
        for (int kk = 0; kk < 256; kk += 4) {
            const v2f a =
                *(const v2f*)&tile[(tw * 16 + m16) * LDSS + kbase + kk + 2 * hi];
            const v2f bq = *(const v2f*)&q[kbase + kk + 2 * hi];
            cacc = __builtin_amdgcn_wmma_f32_16x16x4_f32(
                       false, a, false, bq, (short)0, cacc, false, false);
        }
        // C layout: vgpr v, lane 0 -> M=v; lane 16 -> M=v+8 (any column)
        if (m16 == 0) {
            #pragma unroll
            for (int v = 0; v < 8; ++v) psc[wave * 16 + hi * 8 + v] = cacc[v];
        }
        __syncthreads();

        // --- 3) combine K-quarters, apply scale + mask
        if (t < CHUNK) {
            const int ttw = t >> 4, r = t & 15;
            float raw = 0.f;
            #pragma unroll
            for (int k4 = 0; k4 < 4; ++k4) raw += psc[(2 * k4 + ttw) * 16 + r];
            const float mk = mask[r0 + t];
            msk[t] = mk;
            sc[t]  = (mk < 0.5f) ? NEGVAL : raw * SCALE;
        }
        __syncthreads();

        // chunk max (broadcast LDS reads; every thread gets identical result)
        float m_c = -INFINITY;
        for (int r = 0; r < CHUNK; ++r) m_c = fmaxf(m_c, sc[r]);
        const float newm = fmaxf(m_run, m_c);
        const float fo   = __expf(m_run - newm);   // first chunk: exp(-inf)=0

        if (t < CHUNK) wv[t] = __expf(sc[t] - newm) * msk[t];
        __syncthreads();

        float Z_c = 0.f, S_c = 0.f;
        for (int r = 0; r < CHUNK; ++r) {
            Z_c += __expf(sc[r] - newm);
            S_c += wv[r];
        }
        m_run = newm;
        Z_run = Z_run * fo + Z_c;
        S_run = S_run * fo + S_c;

        // --- 4) weighted accumulation from the SAME LDS tile (no HBM re-read)
        acc.x *= fo; acc.y *= fo; acc.z *= fo; acc.w *= fo;
        #pragma unroll 4
        for (int r = 0; r < CHUNK; ++r) {
            const float  wr = wv[r];
            const float4 v  = *(const float4*)&tile[r * LDSS + 4 * t];
            acc.x += wr * v.x; acc.y += wr * v.y;
            acc.z += wr * v.z; acc.w += wr * v.w;
        }
    }

    ((float4*)(pacc + (size_t)blockIdx.x * DIM))[t] = acc;
    if (t == 0) {
        hdr[blockIdx.x * 3 + 0] = m_run;
        hdr[blockIdx.x * 3 + 1] = Z_run;
        hdr[blockIdx.x * 3 + 2] = S_run;
    }
}

// ---------------------------------------------------------------------------
// Deterministic merge of the NBLK partials per batch:
//   out = sum_p exp(m_p - m) * acc_p   /  (S + 1e-8 * Z)
// with S = sum_p e_p*S_p, Z = sum_p e_p*Z_p  (matches reference semantics)
// ---------------------------------------------------------------------------
__global__ __launch_bounds__(256)
void mp_finalize(const float* __restrict__ hdr, const float* __restrict__ pacc,
                 float* __restrict__ out) {
    const int b = blockIdx.x;
    const int t = threadIdx.x;

    float m = -INFINITY;
    #pragma unroll
    for (int p = 0; p < NBLK; ++p)
        m = fmaxf(m, hdr[(b * NBLK + p) * 3]);

    float Zg = 0.f, Sg = 0.f;
    #pragma unroll
    for (int p = 0; p < NBLK; ++p) {
        const float e = __expf(hdr[(b * NBLK + p) * 3] - m);
        Zg += hdr[(b * NBLK + p) * 3 + 1] * e;
        Sg += hdr[(b * NBLK + p) * 3 + 2] * e;
    }
    const float inv = 1.f / (Sg + 1e-8f * Zg);

    float4 acc = {0.f, 0.f, 0.f, 0.f};
    for (int p = 0; p < NBLK; ++p) {
        const float  e = __expf(hdr[(b * NBLK + p) * 3] - m);
        const float4 v =
            ((const float4*)(pacc + (size_t)(b * NBLK + p) * DIM))[t];
        acc.x += e * v.x; acc.y += e * v.y; acc.z += e * v.z; acc.w += e * v.w;
    }
    float4 o;
    o.x = acc.x * inv; o.y = acc.y * inv; o.z = acc.z * inv; o.w = acc.w * inv;
    ((float4*)(out + (size_t)b * DIM))[t] = o;
}

// ---------------------------------------------------------------------------
extern "C" void kernel_launch(void* const* d_in, const int* in_sizes, int n_in,
                              void* d_out, int out_size, void* d_ws, size_t ws_size,
                              hipStream_t stream) {
    (void)in_sizes; (void)n_in; (void)out_size; (void)ws_size;
    const float* tokens = (const float*)d_in[0];   // [B, N, D] f32
    const float* mask   = (const float*)d_in[1];   // [B, N]    f32
    const float* query  = (const float*)d_in[2];   // [D]       f32
    float* out = (float*)d_out;                    // [B, D]    f32

    float* ws   = (float*)d_ws;
    float* hdr  = ws;                              // 512*3 floats (from 1024 on overlaps pacc)
    float* pacc = ws + 1024;                       // 512*1024 floats (2 MB),
                                                   // 4KB offset keeps 16B align

    mp_fused<<<BATCH * NBLK, 256, 0, stream>>>(tokens, mask, query, hdr, pacc);
    mp_finalize<<<BATCH, 256, 0, stream>>>(hdr, pacc, out);
}